// Phase_15564961481486
// MI455X (gfx1250) — compile-verified
//
#include <hip/hip_runtime.h>

typedef __attribute__((ext_vector_type(2))) float v2f;
typedef __attribute__((ext_vector_type(8))) float v8f;

struct alignas(8) F2 { float x, y; };

// x: (16,32,512,512) f32; weight: (4,1,2,2) f32; out: (16,128,256,256) f32
// out[b, c*4+p, h, w] = sum_{i,j} x[b, c, 2h+i, 2w+j] * weight[p,0,i,j]
//
// V_WMMA_F32_16X16X4_F32: D(16x16) = A(16x4) x B(4x16), K=4 = the 2x2 taps.
//   A rows 0-3 and 8-11 hold weight rows p=0..3 (duplicated), others zero.
//   B columns n = 16 consecutive output w positions; per-lane float2 loads are
//   128B-coalesced per input row. Two WMMAs cover 32 consecutive w outputs; the
//   duplicated A rows place the 2nd group's outputs in lanes 16-31 so each
//   phase plane is written with one full-wave contiguous 128B store.
//
// Each wave processes one full output row (256 outputs = 8 groups), fully
// unrolled: A and both base addresses are computed once, all loads/stores use
// immediate offsets -> ~12 instructions per 1KB of HBM traffic, keeping the
// kernel bandwidth-bound (23.3 TB/s => ~46us for 1.07 GB) instead of
// issue-bound.

__global__ __launch_bounds__(256) void pixel_unshuffle_wmma(
    const float* __restrict__ x, const float* __restrict__ weight,
    float* __restrict__ out) {
  const int lane = threadIdx.x & 31;
  const int wave = threadIdx.x >> 5;
  const unsigned gw = blockIdx.x * 8u + wave;   // one output row per wave

  const int h  = gw & 255;            // output row
  const int bc = gw >> 8;             // b*C + c, 0..511

  // ---- A matrix (weights), built once ----
  // f32 A 16x4 layout: lane%16 = M; lanes 0-15 hold K=0(V0),1(V1); lanes 16-31 K=2,3.
  const int m     = lane & 15;
  const int khalf = (lane >> 4) << 1;           // 0 or 2
  const float* wp = weight + ((m & 3) << 2) + khalf;
  const float wa0 = wp[0];
  const float wa1 = wp[1];
  const bool mval = (m & 4) == 0;               // rows 0-3 and 8-11 carry weights
  v2f A;
  A.x = mval ? wa0 : 0.0f;
  A.y = mval ? wa1 : 0.0f;

  // ---- base addresses, computed once ----
  // lane-half selects tap row i = lane>>4; float2 supplies taps j=0,1 (K pair).
  const int i = lane >> 4;
  const float* xb = x + ((size_t)bc * 512 + 2 * h + i) * 512 + ((lane & 15) << 1);
  float* ob = out + (((size_t)bc * 4) * 256 + h) * 256 + lane;

  const v8f c = {};

  #pragma unroll
  for (int g = 0; g < 8; ++g) {
    const F2 p0 = *(const F2*)(xb + g * 64);        // outputs g*32    .. g*32+15
    const F2 p1 = *(const F2*)(xb + g * 64 + 32);   // outputs g*32+16 .. g*32+31

    v2f B1; B1.x = p0.x; B1.y = p0.y;
    v2f B2; B2.x = p1.x; B2.y = p1.y;

    const v8f d1 = __builtin_amdgcn_wmma_f32_16x16x4_f32(false, A, false, B1,
                                                         (short)0, c, false, false);
    const v8f d2 = __builtin_amdgcn_wmma_f32_16x16x4_f32(false, A, false, B2,
                                                         (short)0, c, false, false);

    // D vgpr p: lanes 0-15 = (M=p, N=lane) -> group even outputs;
    //           lanes 16-31 = (M=p+8, N=lane-16) -> next group (dup A rows).
    #pragma unroll
    for (int p = 0; p < 4; ++p) {
      const float val = (lane < 16) ? d1[p] : d2[p];
      ob[(size_t)p * 65536 + g * 32] = val;         // plane stride 256*256
    }
  }
}

extern "C" void kernel_launch(void* const* d_in, const int* in_sizes, int n_in,
                              void* d_out, int out_size, void* d_ws, size_t ws_size,
                              hipStream_t stream) {
  (void)in_sizes; (void)n_in; (void)out_size; (void)d_ws; (void)ws_size;
  const float* x = (const float*)d_in[0];
  const float* w = (const float*)d_in[1];
  float* out    = (float*)d_out;

  // 512 (b*c) * 256 (h) rows = 131,072 waves; 8 waves per 256-thread block.
  dim3 block(256);
  dim3 grid(16384);
  pixel_unshuffle_wmma<<<grid, block, 0, stream>>>(x, w, out);
}